// ECELoss_71949292142825
// MI455X (gfx1250) — compile-verified
//
#include <hip/hip_runtime.h>
#include <hip/hip_bf16.h>

#define NBINS 15

typedef __attribute__((ext_vector_type(2))) float v2f;
typedef __attribute__((ext_vector_type(8))) float v8f;

// 12-byte sample record (3 class logits), 4-byte aligned
struct __attribute__((aligned(4))) F3 { float x, y, z; };

// ---------------------------------------------------------------------------
// Zero the 32-float accumulator region in workspace (runs every launch so
// graph replays are deterministic; harness does not re-poison between runs).
// ---------------------------------------------------------------------------
__global__ void ece_init_ws(float* __restrict__ ws) {
    if (threadIdx.x < 32) ws[threadIdx.x] = 0.0f;
}

// ---------------------------------------------------------------------------
// Main streaming kernel: per-sample softmax-max + bin, per-wave LDS histogram
// (ds_add_f32), block reduction of the 8 wave-histograms with
// V_WMMA_F32_16X16X4_F32 (ones-vector x matrix = column sums), then one
// global_atomic_add_f32 per bin per block.
// ws layout: [0..14] = sum(conf) per bin, [16..30] = sum(acc) per bin.
// ---------------------------------------------------------------------------
__global__ __launch_bounds__(256) void ece_partial_kernel(
        const float* __restrict__ logits,
        const int*   __restrict__ labels,
        float*       __restrict__ ws,
        int n) {
    __shared__ float hist[8][32];   // [wave][col]; cols 0-14 conf, 16-30 acc

    const int wave = threadIdx.x >> 5;
    const int lane = threadIdx.x & 31;

    hist[wave][lane] = 0.0f;
    __syncthreads();

    const int stride = gridDim.x * blockDim.x;
    const F3* __restrict__ lg = (const F3*)logits;

    for (int i = blockIdx.x * blockDim.x + threadIdx.x; i < n; i += stride) {
        F3 v = lg[i];                 // coalesced 12B/lane stream
        int lab = labels[i];

        // argmax (first-max semantics to match jnp.argmax) + max
        float m = v.x; int pred = 0;
        if (v.y > m) { m = v.y; pred = 1; }
        if (v.z > m) { m = v.z; pred = 2; }

        // confidence = softmax max = 1 / sum(exp(xi - m))
        float s = __expf(v.x - m) + __expf(v.y - m) + __expf(v.z - m);
        float conf = __builtin_amdgcn_rcpf(s);

        // bin: (b[k], b[k+1]] semantics -> ceil(conf*15)-1, clamped
        int bin = (int)ceilf(conf * (float)NBINS) - 1;
        bin = bin < 0 ? 0 : (bin > NBINS - 1 ? NBINS - 1 : bin);

        atomicAdd(&hist[wave][bin], conf);          // ds_add_f32
        if (pred == lab) atomicAdd(&hist[wave][16 + bin], 1.0f);
    }
    __syncthreads();

    // ---- block reduction: column sums of the 8x32 LDS matrix via WMMA ----
    // A = 16x4 all-ones => every row of D = colsum over the 4 K-rows of B.
    // Two chained 16x16x4 WMMAs cover 8 wave-rows; done twice for the two
    // 16-column halves (conf cols 0-15, acc cols 16-31). Wave 0 only
    // (wave-uniform branch => EXEC all-ones inside the wave, as WMMA requires).
    if (threadIdx.x < 32) {
        const int nn = lane & 15;       // column within half
        const int kk = lane >> 4;       // K-row parity per lane-half layout

        v2f a; a.x = 1.0f; a.y = 1.0f;  // all-ones A (layout-independent)

        v2f b0, b1, b2, b3;
        b0.x = hist[kk][nn];        b0.y = hist[kk + 2][nn];        // waves 0-3, cols 0-15
        b1.x = hist[kk + 4][nn];    b1.y = hist[kk + 6][nn];        // waves 4-7, cols 0-15
        b2.x = hist[kk][16 + nn];   b2.y = hist[kk + 2][16 + nn];   // waves 0-3, cols 16-31
        b3.x = hist[kk + 4][16 + nn]; b3.y = hist[kk + 6][16 + nn]; // waves 4-7, cols 16-31

        v8f c0 = {};
        c0 = __builtin_amdgcn_wmma_f32_16x16x4_f32(false, a, false, b0,
                                                   (short)0, c0, false, false);
        c0 = __builtin_amdgcn_wmma_f32_16x16x4_f32(false, a, false, b1,
                                                   (short)0, c0, false, false);
        v8f c1 = {};
        c1 = __builtin_amdgcn_wmma_f32_16x16x4_f32(false, a, false, b2,
                                                   (short)0, c1, false, false);
        c1 = __builtin_amdgcn_wmma_f32_16x16x4_f32(false, a, false, b3,
                                                   (short)0, c1, false, false);

        // D layout: VGPR0, lanes 0-15 hold row M=0, N=lane -> column sums.
        if (lane < NBINS) {
            unsafeAtomicAdd(&ws[lane],      c0[0]);  // global_atomic_add_f32
            unsafeAtomicAdd(&ws[16 + lane], c1[0]);
        }
    }
}

// ---------------------------------------------------------------------------
// Finalize: ece = sum_b |sum_conf[b] - sum_acc[b]| / N
// ---------------------------------------------------------------------------
__global__ void ece_finalize_kernel(const float* __restrict__ ws,
                                    float* __restrict__ out, float inv_n) {
    if (threadIdx.x == 0) {
        float s = 0.0f;
        #pragma unroll
        for (int i = 0; i < NBINS; ++i)
            s += fabsf(ws[i] - ws[16 + i]);
        out[0] = s * inv_n;
    }
}

extern "C" void kernel_launch(void* const* d_in, const int* in_sizes, int n_in,
                              void* d_out, int out_size, void* d_ws, size_t ws_size,
                              hipStream_t stream) {
    const float* logits = (const float*)d_in[0];  // [N,3] f32
    const int*   labels = (const int*)d_in[1];    // [N]   i32
    const int n = in_sizes[1];                    // N samples
    float* ws  = (float*)d_ws;
    float* out = (float*)d_out;

    ece_init_ws<<<1, 64, 0, stream>>>(ws);

    const int threads = 256;
    const int blocks  = 4096;   // 1M threads -> 16 samples/thread grid-stride
    ece_partial_kernel<<<blocks, threads, 0, stream>>>(logits, labels, ws, n);

    ece_finalize_kernel<<<1, 32, 0, stream>>>(ws, out, 1.0f / (float)n);
}